// RNN_197568496340
// MI455X (gfx1250) — compile-verified
//
#include <hip/hip_runtime.h>
#include <hip/hip_bf16.h>

typedef __attribute__((ext_vector_type(2))) float v2f;
typedef __attribute__((ext_vector_type(8))) float v8f;

#define T_  1024
#define B_  2048
#define IN_ 2
#define H_  20
#define L_  6
#define C_  2
#define NROW  (T_ * B_)      // 2,097,152 flattened (t,b) rows
#define NTILE (NROW / 16)    // 131,072 row tiles

// relu without the compiler's NaN-canonicalize double-max (inputs are finite)
static __device__ __forceinline__ float relu1(float x) {
    float r;
    asm("v_max_num_f32_e32 %0, 0, %1" : "=v"(r) : "v"(x));
    return r;
}

// ---------------------------------------------------------------------------
// k_proj<IN_DIM>: xb[row, 0..19] = in[row, 0..IN_DIM-1] @ Wih^T + (bih + bhh)
// Parallel GEMM, M-tiles of 16 rows, N padded 20->32 (2 tiles), K in chunks
// of 4 via V_WMMA_F32_16X16X4_F32. IN_DIM is compile-time so all register
// array indexing is static.
// ---------------------------------------------------------------------------
template <int IN_DIM>
__global__ void k_proj(const float* __restrict__ in,
                       const float* __restrict__ Wih,   // [H_, IN_DIM]
                       const float* __restrict__ bih,
                       const float* __restrict__ bhh,
                       float* __restrict__ xb)          // [NROW, H_]
{
    constexpr int NCH = (IN_DIM + 3) / 4;               // 1 (layer 0) or 5
    const int lane = threadIdx.x & 31;
    const int half = lane >> 4;       // A: k+2*half ; C/D: rows +8*half
    const int lm   = lane & 15;
    const int wid  = (blockIdx.x * blockDim.x + threadIdx.x) >> 5;
    const int nw   = (gridDim.x * blockDim.x) >> 5;

    // Preload B operand: B[k][n] = Wih[n][k]  (n = output col, padded to 32)
    v2f Bt[2][NCH];
    float bias[2];
#pragma unroll
    for (int tn = 0; tn < 2; ++tn) {
        const int j = tn * 16 + lm;
#pragma unroll
        for (int c = 0; c < NCH; ++c) {
            const int k = c * 4 + half * 2;
            v2f b; b.x = 0.f; b.y = 0.f;
            if (j < H_ && k + 1 < IN_DIM)
                b = *(const v2f*)(Wih + j * IN_DIM + k);   // global_load_b64
            Bt[tn][c] = b;
        }
        bias[tn] = (j < H_) ? (bih[j] + bhh[j]) : 0.f;
    }

    for (int tile = wid; tile < NTILE; tile += nw) {
        const long long row = (long long)tile * 16 + lm;   // A row m = lm
        v2f A[NCH];
        if (IN_DIM == 2) {
            // Unconditional b64 load of cols 0..1; half==1 lanes (k=2,3)
            // need zeros -> branchless cndmask instead of exec-branch.
            v2f a = *(const v2f*)(in + row * 2);
            if (half) { a.x = 0.f; a.y = 0.f; }
            A[0] = a;
        } else {
#pragma unroll
            for (int c = 0; c < NCH; ++c) {
                const int k = c * 4 + half * 2;
                A[c] = *(const v2f*)(in + row * IN_DIM + k);  // k+1 <= 19
            }
        }
#pragma unroll
        for (int tn = 0; tn < 2; ++tn) {
            v8f c;
#pragma unroll
            for (int v = 0; v < 8; ++v) c[v] = bias[tn];
#pragma unroll
            for (int ch = 0; ch < NCH; ++ch)
                c = __builtin_amdgcn_wmma_f32_16x16x4_f32(
                        false, A[ch], false, Bt[tn][ch], (short)0, c, false, false);
            const long long r0 = (long long)tile * 16;
            const int n = tn * 16 + lm;
            if (n < H_) {                       // single exec region, 8 stores
#pragma unroll
                for (int v = 0; v < 8; ++v) {
                    const int m = v + half * 8;            // C/D layout row
                    xb[(r0 + m) * H_ + n] = c[v];
                }
            }
        }
    }
}

// ---------------------------------------------------------------------------
// k_scan: h_t = relu(xb_t + h_{t-1} @ Whh^T), in place on xb (read tile t,
// overwrite tile t). One wave owns one 16-row batch tile for all T steps.
// h round-trips through a double-buffered LDS region to convert WMMA D-layout
// back into A-layout for the next step (stride-20 is bank-conflict-free).
// The xb (C-operand) loads are software-pipelined one step ahead so global
// load latency overlaps the previous step's compute — only 128 waves run, so
// nothing else hides it.
// ---------------------------------------------------------------------------
__global__ void k_scan(const float* __restrict__ Whh,   // [H_, H_] (this layer)
                       float* __restrict__ xb)          // [NROW, H_] in/out
{
    __shared__ float hbuf[2][16 * H_];
    const int lane = threadIdx.x & 31;
    const int half = lane >> 4;
    const int lm   = lane & 15;
    const int b0   = blockIdx.x * 16;

    // Preload Whh^T as B operand: B[k][j] = Whh[j][k], j padded to 32.
    v2f Bt[2][5];
#pragma unroll
    for (int tn = 0; tn < 2; ++tn) {
        const int j = tn * 16 + lm;
#pragma unroll
        for (int c = 0; c < 5; ++c) {
            const int k = c * 4 + half * 2;
            v2f b; b.x = 0.f; b.y = 0.f;
            if (j < H_)
                b = *(const v2f*)(Whh + j * H_ + k);    // k+1 <= 19 always
            Bt[tn][c] = b;
        }
    }

    // h_{-1} = 0
#pragma unroll
    for (int v = 0; v < 8; ++v) {
        const int m = v + half * 8;
        hbuf[0][m * H_ + lm] = 0.f;
        if (lm < 4) hbuf[0][m * H_ + 16 + lm] = 0.f;
    }
    __syncthreads();   // single-wave workgroup: barrier -> S_NOP, acts as fence

    // Column offsets: tile0 = lm (all valid); tile1 = 16+lm but only lm<4
    // valid -> C loads use a clamped in-bounds column (pad cols of the
    // accumulator only feed dead output columns, value irrelevant).
    const int n1c = 16 + (lm < 4 ? lm : 3);   // clamped col for tile1 C loads

    // Prefetch C(t=0)
    v8f p0, p1;
#pragma unroll
    for (int v = 0; v < 8; ++v) {
        const int m = v + half * 8;
        p0[v] = xb[((long long)b0 + m) * H_ + lm];
        p1[v] = xb[((long long)b0 + m) * H_ + n1c];
    }

    for (int t = 0; t < T_; ++t) {
        const int cur = t & 1, nxt = cur ^ 1;
        const long long r0 = (long long)t * B_ + b0;
        // Next step's row base (clamped at t = T-1: dead re-read, value unused)
        const long long rn = (t + 1 < T_) ? (r0 + B_) : r0;

        v8f c0 = p0, c1 = p1;

        // Prefetch C(t+1) — overlaps with this step's WMMA chain
#pragma unroll
        for (int v = 0; v < 8; ++v) {
            const int m = v + half * 8;
            p0[v] = xb[(rn + m) * H_ + lm];
            p1[v] = xb[(rn + m) * H_ + n1c];
        }

        // A = h_{t-1} in A-layout from LDS (ds_load_2addr_b64, conflict-free)
        v2f A[5];
#pragma unroll
        for (int c = 0; c < 5; ++c) {
            const int k = c * 4 + half * 2;
            A[c] = *(const v2f*)(&hbuf[cur][lm * H_ + k]);
        }

#pragma unroll
        for (int ch = 0; ch < 5; ++ch) {
            c0 = __builtin_amdgcn_wmma_f32_16x16x4_f32(
                    false, A[ch], false, Bt[0][ch], (short)0, c0, false, false);
            c1 = __builtin_amdgcn_wmma_f32_16x16x4_f32(
                    false, A[ch], false, Bt[1][ch], (short)0, c1, false, false);
        }

        // relu (single v_max_num each, no canonicalize)
        float h0[8], h1[8];
#pragma unroll
        for (int v = 0; v < 8; ++v) {
            h0[v] = relu1(c0[v]);
            h1[v] = relu1(c1[v]);
        }

        // tile0: all 16 cols valid -> unconditional stores (LDS + global)
#pragma unroll
        for (int v = 0; v < 8; ++v) {
            const int m = v + half * 8;
            hbuf[nxt][m * H_ + lm] = h0[v];
            xb[(r0 + m) * H_ + lm] = h0[v];
        }
        // tile1: only cols 16..19 valid -> one predicated region
        if (lm < 4) {
            const int n1 = 16 + lm;
#pragma unroll
            for (int v = 0; v < 8; ++v) {
                const int m = v + half * 8;
                hbuf[nxt][m * H_ + n1] = h1[v];
                xb[(r0 + m) * H_ + n1] = h1[v];
            }
        }
        __syncthreads();   // order LDS stores (t) before LDS loads (t+1)
    }
}

// ---------------------------------------------------------------------------
// k_head: logits = h @ fc_w^T + fc_b, then log_softmax over C=2.
// ---------------------------------------------------------------------------
__global__ void k_head(const float* __restrict__ h,
                       const float* __restrict__ fc_w,   // [C_, H_]
                       const float* __restrict__ fc_b,   // [C_]
                       float* __restrict__ out)          // [NROW, C_]
{
    const long long i = (long long)blockIdx.x * blockDim.x + threadIdx.x;
    if (i >= (long long)NROW) return;
    const float* hp = h + i * H_;
    float l0 = fc_b[0], l1 = fc_b[1];
#pragma unroll
    for (int k = 0; k < H_; ++k) {
        const float v = hp[k];
        l0 += v * fc_w[k];
        l1 += v * fc_w[H_ + k];
    }
    const float m   = fmaxf(l0, l1);
    const float lse = m + logf(expf(l0 - m) + expf(l1 - m));
    out[i * 2 + 0] = l0 - lse;
    out[i * 2 + 1] = l1 - lse;
}

// ---------------------------------------------------------------------------
extern "C" void kernel_launch(void* const* d_in, const int* in_sizes, int n_in,
                              void* d_out, int out_size, void* d_ws, size_t ws_size,
                              hipStream_t stream) {
    const float* x     = (const float*)d_in[0];  // [T,B,IN]
    const float* W_ih0 = (const float*)d_in[1];  // [H,IN]
    const float* W_ih  = (const float*)d_in[2];  // [L-1,H,H]
    const float* W_hh  = (const float*)d_in[3];  // [L,H,H]
    const float* b_ih  = (const float*)d_in[4];  // [L,H]
    const float* b_hh  = (const float*)d_in[5];  // [L,H]
    const float* fc_w  = (const float*)d_in[6];  // [C,H]
    const float* fc_b  = (const float*)d_in[7];  // [C]
    float* out = (float*)d_out;

    // Ping-pong activation buffers: 2 x (T*B*H) floats = 2 x ~168 MB
    float* bufA = (float*)d_ws;
    float* bufB = bufA + (size_t)NROW * H_;

    const float* cur_in  = x;
    float*       cur_buf = bufA;
    float*       other   = bufB;

    for (int l = 0; l < L_; ++l) {
        const float* bi = b_ih + (size_t)l * H_;
        const float* bh = b_hh + (size_t)l * H_;
        if (l == 0) {
            k_proj<IN_><<<1024, 256, 0, stream>>>(cur_in, W_ih0, bi, bh, cur_buf);
        } else {
            k_proj<H_><<<1024, 256, 0, stream>>>(cur_in,
                                                 W_ih + (size_t)(l - 1) * H_ * H_,
                                                 bi, bh, cur_buf);
        }
        k_scan<<<B_ / 16, 32, 0, stream>>>(W_hh + (size_t)l * H_ * H_, cur_buf);
        cur_in = cur_buf;
        float* tmp = other; other = cur_buf; cur_buf = tmp;
    }
    k_head<<<(NROW + 255) / 256, 256, 0, stream>>>(cur_in, fc_w, fc_b, out);
}